// Proxy_Mixer_70901320123307
// MI455X (gfx1250) — compile-verified
//
#include <hip/hip_runtime.h>
#include <hip/hip_bf16.h>

typedef float v2f __attribute__((ext_vector_type(2)));
typedef float v8f __attribute__((ext_vector_type(8)));

#define BB 4
#define SS 2048
#define DD 512
#define PP 16
#define CC 64           // chunk length (s)
#define NC (SS / CC)    // 32 chunks

// D = A(16x4) * B(4x16) + C, fp32 in/out
__device__ __forceinline__ v8f wmma_f32(v2f a, v2f b, v8f c) {
  return __builtin_amdgcn_wmma_f32_16x16x4_f32(false, a, false, b, (short)0, c,
                                               false, false);
}

// ---------------------------------------------------------------------------
// Kernel 1: per (b, chunk): P_c = proxy @ X_c^T  (16 x 64, K=512)
//                           dA  = P_c @ X_c      (16 x 512, K=64)
// ---------------------------------------------------------------------------
__global__ __launch_bounds__(128) void k1_chunk(const float* __restrict__ x,
                                                const float* __restrict__ proxy,
                                                float* __restrict__ Pws,
                                                float* __restrict__ dA) {
  __shared__ float sP[PP][CC + 4];

  const int bc = blockIdx.x;
  const int b = bc / NC, c = bc % NC;
  const int tid = threadIdx.x;
  const int w = tid >> 5;           // wave id 0..3
  const int l = tid & 31;
  const int lane16 = l & 15;
  const int half = l >> 4;
  const int kb = half * 2;          // K base within a 4-wide slab
  const int cbase = c * CC;

  const float* xb = x + (size_t)b * SS * DD;

  // ---- P tile: wave w owns s-columns [16*w, 16*w+16) of this chunk --------
  {
    v8f acc = {};
    const float* arow = proxy + (size_t)lane16 * DD;            // A: m = lane16
    const float* brow = xb + (size_t)(cbase + 16 * w + lane16) * DD; // B: n = lane16
    for (int kk = 0; kk < DD / 4; ++kk) {
      const int k = kk * 4 + kb;
      v2f a, bf;
      a.x = arow[k];     a.y = arow[k + 1];
      bf.x = brow[k];    bf.y = brow[k + 1];
      acc = wmma_f32(a, bf, acc);
    }
#pragma unroll
    for (int j = 0; j < 8; ++j) {
      const int m = j + 8 * half;                 // proxy index
      const int s = 16 * w + lane16;              // s within chunk
      sP[m][s] = acc[j];
      Pws[((size_t)b * PP + m) * SS + cbase + s] = acc[j];
    }
  }
  __syncthreads();

  // ---- dA = sP(16x64) @ X(64x512): wave w owns d-tiles w, w+4, ... --------
  float* dAp = dA + (size_t)bc * PP * DD;
  for (int dt = w; dt < DD / 16; dt += 4) {
    const int d0 = dt * 16;
    v8f acc = {};
    for (int kk = 0; kk < CC / 4; ++kk) {
      const int k = kk * 4 + kb;
      v2f a, bf;
      a.x = sP[lane16][k];  a.y = sP[lane16][k + 1];
      const float* xr = xb + (size_t)(cbase + k) * DD + d0 + lane16;
      bf.x = xr[0];         bf.y = xr[DD];
      acc = wmma_f32(a, bf, acc);
    }
#pragma unroll
    for (int j = 0; j < 8; ++j) {
      const int m = j + 8 * half;
      dAp[(size_t)m * DD + d0 + lane16] = acc[j];
    }
  }
}

// ---------------------------------------------------------------------------
// Kernel 2: exclusive prefix of dA over chunks -> Abase. One block per batch;
// each thread keeps 32 state elements in registers (16*512/256).
// ---------------------------------------------------------------------------
__global__ __launch_bounds__(256) void k2_prefix(const float* __restrict__ dA,
                                                 float* __restrict__ Abase) {
  const int b = blockIdx.x;
  const int tid = threadIdx.x;
  float acc[32];
#pragma unroll
  for (int i = 0; i < 32; ++i) acc[i] = 0.f;
  for (int c = 0; c < NC; ++c) {
    const float* src = dA + ((size_t)(b * NC + c)) * (PP * DD);
    float* dst = Abase + ((size_t)(b * NC + c)) * (PP * DD);
#pragma unroll
    for (int i = 0; i < 32; ++i) {
      const int e = tid + i * 256;
      dst[e] = acc[i];
      acc[i] += src[e];
    }
  }
}

// ---------------------------------------------------------------------------
// Kernel 3: per (b, chunk): W = softmax_p(P/4);  L = tril(W^T P);
//           out = W^T @ Abase + L @ X   (all WMMA f32 16x16x4)
// ---------------------------------------------------------------------------
__global__ __launch_bounds__(128) void k3_out(const float* __restrict__ x,
                                              const float* __restrict__ Pws,
                                              const float* __restrict__ Abase,
                                              float* __restrict__ out) {
  __shared__ float sPc[PP][CC + 4];
  __shared__ float sW[PP][CC + 4];
  __shared__ float sL[4][16][CC + 4];

  const int bc = blockIdx.x;
  const int b = bc / NC, c = bc % NC;
  const int tid = threadIdx.x;
  const int w = tid >> 5;
  const int l = tid & 31;
  const int lane16 = l & 15;
  const int half = l >> 4;
  const int kb = half * 2;
  const int cbase = c * CC;

  // stage P chunk
  for (int i = tid; i < PP * CC; i += 128) {
    const int p = i >> 6, s = i & (CC - 1);
    sPc[p][s] = Pws[((size_t)b * PP + p) * SS + cbase + s];
  }
  __syncthreads();

  // softmax over proxies, one column per thread (scale 1/sqrt(16) = 0.25)
  if (tid < CC) {
    float v[PP], vmax = -3.4e38f;
#pragma unroll
    for (int p = 0; p < PP; ++p) { v[p] = sPc[p][tid] * 0.25f; vmax = fmaxf(vmax, v[p]); }
    float sum = 0.f;
#pragma unroll
    for (int p = 0; p < PP; ++p) { v[p] = __expf(v[p] - vmax); sum += v[p]; }
    const float inv = 1.f / sum;
#pragma unroll
    for (int p = 0; p < PP; ++p) sW[p][tid] = v[p] * inv;
  }
  __syncthreads();

  // ---- Phase A: L tiles for this wave's s-tile (st = w) -------------------
  const int st = w;
  const int s0 = st * 16;
  for (int tt = 0; tt <= st; ++tt) {
    const int t0 = tt * 16;
    v8f accL = {};
    for (int kk = 0; kk < 4; ++kk) {        // K = proxies = 16
      const int k = kk * 4 + kb;
      v2f a, bf;
      a.x = sW[k][s0 + lane16];     a.y = sW[k + 1][s0 + lane16];   // A = W^T
      bf.x = sPc[k][t0 + lane16];   bf.y = sPc[k + 1][t0 + lane16]; // B = P
      accL = wmma_f32(a, bf, accL);
    }
#pragma unroll
    for (int j = 0; j < 8; ++j) {
      const int m = j + 8 * half;           // row s (local to tile)
      float val = accL[j];
      if (tt == st && (t0 + lane16) > (s0 + m)) val = 0.f;  // causal (inclusive)
      sL[w][m][t0 + lane16] = val;
    }
  }
  __syncthreads();

  // ---- Phase B: out = W^T @ Abase + L @ X ---------------------------------
  const float* Ab = Abase + (size_t)bc * PP * DD;
  const float* xb = x + (size_t)b * SS * DD;
  float* outb = out + (size_t)b * SS * DD;

  for (int dt = 0; dt < DD / 16; ++dt) {
    const int d0 = dt * 16;
    v8f acc = {};
    // carry term: K = proxies = 16
    for (int kk = 0; kk < 4; ++kk) {
      const int k = kk * 4 + kb;
      v2f a, bf;
      a.x = sW[k][s0 + lane16];             a.y = sW[k + 1][s0 + lane16];
      bf.x = Ab[(size_t)k * DD + d0 + lane16];
      bf.y = Ab[(size_t)(k + 1) * DD + d0 + lane16];
      acc = wmma_f32(a, bf, acc);
    }
    // intra-chunk term: K = t up to (st+1)*16
    for (int tt = 0; tt <= st; ++tt) {
      for (int kk = 0; kk < 4; ++kk) {
        const int k = tt * 16 + kk * 4 + kb;
        v2f a, bf;
        a.x = sL[w][lane16][k];   a.y = sL[w][lane16][k + 1];
        const float* xr = xb + (size_t)(cbase + k) * DD + d0 + lane16;
        bf.x = xr[0];             bf.y = xr[DD];
        acc = wmma_f32(a, bf, acc);
      }
    }
#pragma unroll
    for (int j = 0; j < 8; ++j) {
      const int m = j + 8 * half;
      outb[(size_t)(cbase + s0 + m) * DD + d0 + lane16] = acc[j];
    }
  }
}

// ---------------------------------------------------------------------------
extern "C" void kernel_launch(void* const* d_in, const int* in_sizes, int n_in,
                              void* d_out, int out_size, void* d_ws, size_t ws_size,
                              hipStream_t stream) {
  (void)in_sizes; (void)n_in; (void)out_size; (void)ws_size;
  const float* x = (const float*)d_in[0];
  const float* proxy = (const float*)d_in[1];
  float* out = (float*)d_out;

  float* Pws = (float*)d_ws;                       // B*16*S        = 131072 f
  float* dA = Pws + (size_t)BB * PP * SS;          // B*NC*16*512   = 1048576 f
  float* Abase = dA + (size_t)BB * NC * PP * DD;   // B*NC*16*512   = 1048576 f
  // total workspace ~8.7 MB

  k1_chunk<<<dim3(BB * NC), dim3(128), 0, stream>>>(x, proxy, Pws, dA);
  k2_prefix<<<dim3(BB), dim3(256), 0, stream>>>(dA, Abase);
  k3_out<<<dim3(BB * NC), dim3(128), 0, stream>>>(x, Pws, Abase, out);
}